// Distance_89687507075609
// MI455X (gfx1250) — compile-verified
//
#include <hip/hip_runtime.h>
#include <math.h>

// Radius graph for MI455X (gfx1250, wave32):
//  - one workgroup (256 threads = 8 waves) per 128-atom molecule
//  - Gram tiles via V_WMMA_F32_16X16X4_F32 (inner dim 3, padded to 4)
//  - d2 staged in LDS (64 rows x 132 stride, two halves)
//  - per-row K=32 nearest selection via wave32 shfl_xor argmin (lane k writes edge k)

typedef __attribute__((ext_vector_type(2))) float v2f;
typedef __attribute__((ext_vector_type(8))) float v8f;

#define ATOMS   128
#define KSEL    32
#define STRIDE  132          // 128 cols + pad; keeps WMMA epilogue stores conflict-free
#define R2      25.0f        // CUTOFF^2

__global__ __launch_bounds__(256) void radius_graph_wmma_kernel(
    const float* __restrict__ pos,
    int* __restrict__ e_src, int* __restrict__ e_dst,
    float* __restrict__ e_w, float* __restrict__ e_vec)
{
    __shared__ float sx[ATOMS], sy[ATOMS], sz[ATOMS], ssq[ATOMS];
    __shared__ float d2buf[64 * STRIDE];   // one 64-row half at a time (~33 KB)

    const int mol  = blockIdx.x;
    const int base = mol * ATOMS;
    const int tid  = (int)threadIdx.x;
    const int wave = tid >> 5;
    const int lane = tid & 31;

    // ---- stage positions + squared norms in LDS ----
    if (tid < ATOMS) {
        float x = pos[(base + tid) * 3 + 0];
        float y = pos[(base + tid) * 3 + 1];
        float z = pos[(base + tid) * 3 + 2];
        sx[tid] = x; sy[tid] = y; sz[tid] = z;
        ssq[tid] = x * x + y * y + z * z;
    }
    __syncthreads();

    for (int h = 0; h < 2; ++h) {          // two 64-row halves
        // ================= Gram tiles via WMMA f32 16x16x4 =================
        // 4 row-tiles x 8 col-tiles this half; wave -> (row-tile, 4 col-tiles)
        const int ti_local = wave >> 1;                 // 0..3
        const int Rrow = (h * 4 + ti_local) * 16;       // global row-tile base

        // A fragment: 16x4, lanes 0-15 carry K=0,1 (x,y); lanes 16-31 carry K=2,3 (z,0)
        v2f a;
        if (lane < 16) { a.x = sx[Rrow + lane];      a.y = sy[Rrow + lane]; }
        else           { a.x = sz[Rrow + lane - 16]; a.y = 0.0f;            }

        const int tj0 = (wave & 1) * 4;
        #pragma unroll
        for (int tjj = 0; tjj < 4; ++tjj) {
            const int Ccol = (tj0 + tjj) * 16;
            // B fragment: 4x16, same split across lane halves
            v2f b;
            if (lane < 16) { b.x = sx[Ccol + lane];      b.y = sy[Ccol + lane]; }
            else           { b.x = sz[Ccol + lane - 16]; b.y = 0.0f;            }

            v8f c = {};
            c = __builtin_amdgcn_wmma_f32_16x16x4_f32(
                    false, a, false, b, (short)0, c, false, false);

            // C/D layout: vgpr v, lane l -> row v + 8*(l>>4), col l&15
            const int colN  = Ccol + (lane & 15);
            const int rbase = ti_local * 16 + ((lane >> 4) << 3);
            const float sqc = ssq[colN];
            #pragma unroll
            for (int v = 0; v < 8; ++v) {
                const int rloc = rbase + v;                 // 0..63 within half
                const int Mrow = h * 64 + rloc;             // 0..127 within molecule
                float d2 = ssq[Mrow] + sqc - 2.0f * c[v];
                d2 = fmaxf(d2, 0.0f);
                d2buf[rloc * STRIDE + colN] = (d2 <= R2) ? d2 : __builtin_inff();
            }
        }
        __syncthreads();

        // ================= wave32 top-K selection (K == 32 == wave width) ==
        for (int i = 0; i < 8; ++i) {
            const int rloc = wave * 8 + i;                  // 0..63
            const int rl   = h * 64 + rloc;                 // local row index
            const int rowg = base + rl;                     // global row index

            // lane owns 4 candidate columns: j = lane*4 .. lane*4+3 (b128 LDS read)
            const float4 quad = ((const float4*)&d2buf[rloc * STRIDE])[lane];
            // key = (d2 bits, col): monotone for non-negative f32, ties -> lower col
            unsigned long long k0 = ((unsigned long long)__float_as_uint(quad.x) << 32) | (unsigned)(lane * 4 + 0);
            unsigned long long k1 = ((unsigned long long)__float_as_uint(quad.y) << 32) | (unsigned)(lane * 4 + 1);
            unsigned long long k2 = ((unsigned long long)__float_as_uint(quad.z) << 32) | (unsigned)(lane * 4 + 2);
            unsigned long long k3 = ((unsigned long long)__float_as_uint(quad.w) << 32) | (unsigned)(lane * 4 + 3);

            for (int kk = 0; kk < KSEL; ++kk) {
                unsigned long long m01 = (k0 < k1) ? k0 : k1;
                unsigned long long m23 = (k2 < k3) ? k2 : k3;
                unsigned long long key = (m01 < m23) ? m01 : m23;
                #pragma unroll
                for (int off = 16; off > 0; off >>= 1) {
                    unsigned long long o = __shfl_xor(key, off, 32);
                    key = (o < key) ? o : key;
                }
                const unsigned col  = (unsigned)(key & 0xffffffffu);
                const unsigned bits = (unsigned)(key >> 32);

                // invalidate the winning slot in its owning lane
                if ((col >> 2) == (unsigned)lane) {
                    const unsigned s = col & 3u;
                    if      (s == 0) k0 = ~0ULL;
                    else if (s == 1) k1 = ~0ULL;
                    else if (s == 2) k2 = ~0ULL;
                    else             k3 = ~0ULL;
                }

                // selection kk's edge is written by lane kk (K == 32 lanes)
                if (lane == kk) {
                    const long long e = (long long)rowg * KSEL + kk;
                    int dstg; float vx, vy, vz, w;
                    if (bits >= 0x7f800000u) {              // inf -> padding self-loop
                        dstg = rowg; vx = vy = vz = 0.0f; w = 0.0f;
                    } else {
                        const int cl = (int)col;
                        dstg = base + cl;
                        vx = sx[rl] - sx[cl];
                        vy = sy[rl] - sy[cl];
                        vz = sz[rl] - sz[cl];
                        const float ss2 = vx * vx + vy * vy + vz * vz;
                        w = (rowg != dstg) ? sqrtf(ss2) : 0.0f;
                    }
                    e_src[e] = rowg;
                    e_dst[e] = dstg;
                    e_w[e]   = w;
                    e_vec[e * 3 + 0] = vx;
                    e_vec[e * 3 + 1] = vy;
                    e_vec[e * 3 + 2] = vz;
                }
            }
        }
        __syncthreads();   // d2buf reused by next half
    }
}

extern "C" void kernel_launch(void* const* d_in, const int* in_sizes, int n_in,
                              void* d_out, int out_size, void* d_ws, size_t ws_size,
                              hipStream_t stream)
{
    (void)n_in; (void)out_size; (void)d_ws; (void)ws_size;
    const float* pos = (const float*)d_in[0];   // [N,3] f32
    const int N  = in_sizes[1];                 // batch array length == N (12288)
    const int NK = N * KSEL;

    // outputs concatenated flat in return order:
    // edge_index [2, N*K] (src then dst), edge_weight [N*K], edge_vec [N*K, 3]
    int*   e_src = (int*)d_out;
    int*   e_dst = e_src + NK;
    float* e_w   = (float*)d_out + 2 * (size_t)NK;
    float* e_vec = (float*)d_out + 3 * (size_t)NK;

    const int nmol = N / ATOMS;                 // 96 molecules
    radius_graph_wmma_kernel<<<nmol, 256, 0, stream>>>(pos, e_src, e_dst, e_w, e_vec);
}